// TriangulatorInternal_20315195310084
// MI455X (gfx1250) — compile-verified
//
#include <hip/hip_runtime.h>
#include <hip/hip_bf16.h>
#include <stdint.h>

#define NCAMS 8
#define BLK   256

struct F3 { float x, y, z; };

__device__ __forceinline__ F3 cross3(F3 a, F3 b) {
    return F3{ a.y * b.z - a.z * b.y,
               a.z * b.x - a.x * b.z,
               a.x * b.y - a.y * b.x };
}
__device__ __forceinline__ float dot3(F3 a, F3 b) {
    return a.x * b.x + a.y * b.y + a.z * b.z;
}

// Issue an async global->LDS 8-byte copy for this lane (gfx1250 ASYNCcnt path).
__device__ __forceinline__ void async_copy_b64(unsigned lds_byte_addr, const void* gaddr) {
    asm volatile("global_load_async_to_lds_b64 %0, %1, off"
                 :: "v"(lds_byte_addr), "v"((unsigned long long)(uintptr_t)gaddr)
                 : "memory");
}

__global__ __launch_bounds__(BLK) void triangulate_kernel(
    const float* __restrict__ uv,          // [N,2]
    const float* __restrict__ uvs_others,  // [8,N,2]
    const float* __restrict__ Rt,          // [4,4] row-major
    const float* __restrict__ Rts_others,  // [8,4,4]
    const float* __restrict__ fvec,        // [2]
    const float* __restrict__ centre,      // [2]
    const float* __restrict__ p_max_error, // [1]
    const float* __restrict__ p_min_dis,   // [1]
    float* __restrict__ out,               // 6N floats: kpts3d | depth | best_dis | mask
    int N)
{
    __shared__ float  relM[NCAMS][12];   // per-cam rel_Rt: R row-major [0..8], t [9..11]
    __shared__ float2 stage[2][BLK];     // double-buffered async staging for uv_other

    const int tid = threadIdx.x;

    // ---- per-block setup: rel_Rt[cam] = Rt @ inv(Rts_others[cam]) (rigid inverse) ----
    if (tid < NCAMS) {
        const float* B = Rts_others + tid * 16;
        float AR[3][3], At[3], BR[3][3], Bt[3];
        #pragma unroll
        for (int i = 0; i < 3; ++i) {
            #pragma unroll
            for (int k = 0; k < 3; ++k) { AR[i][k] = Rt[i * 4 + k]; BR[i][k] = B[i * 4 + k]; }
            At[i] = Rt[i * 4 + 3];
            Bt[i] = B[i * 4 + 3];
        }
        float invt[3];
        #pragma unroll
        for (int k = 0; k < 3; ++k)
            invt[k] = -(BR[0][k] * Bt[0] + BR[1][k] * Bt[1] + BR[2][k] * Bt[2]);
        #pragma unroll
        for (int i = 0; i < 3; ++i) {
            #pragma unroll
            for (int j = 0; j < 3; ++j)
                relM[tid][i * 3 + j] = AR[i][0] * BR[j][0] + AR[i][1] * BR[j][1] + AR[i][2] * BR[j][2];
            relM[tid][9 + i] = At[i] + AR[i][0] * invt[0] + AR[i][1] * invt[1] + AR[i][2] * invt[2];
        }
    }
    __syncthreads();

    const long long i      = (long long)blockIdx.x * BLK + tid;
    const long long iclamp = (i < N) ? i : (long long)N - 1;   // tail lanes: valid addr, store predicated

    const float fx = fvec[0], fy = fvec[1];
    const float cx = centre[0], cy = centre[1];
    const float inv_fx = __builtin_amdgcn_rcpf(fx);
    const float inv_fy = __builtin_amdgcn_rcpf(fy);
    const float max_error = p_max_error[0];
    const float min_dis   = p_min_dis[0];

    // d1: unit ray of reference camera (loop-invariant)
    const float2 uv0 = ((const float2*)uv)[iclamp];
    F3 d1 = { (uv0.x - cx) * inv_fx, (uv0.y - cy) * inv_fy, 1.0f };
    {
        const float s = __builtin_amdgcn_rsqf(dot3(d1, d1));
        d1.x *= s; d1.y *= s; d1.z *= s;
    }
    const F3 d1x10 = { d1.x * 10.0f, d1.y * 10.0f, d1.z * 10.0f };

    const unsigned lds_addr[2] = {
        (unsigned)(uintptr_t)(void*)&stage[0][tid],
        (unsigned)(uintptr_t)(void*)&stage[1][tid]
    };
    const float2* uvo_base = (const float2*)uvs_others;

    // prime the pipeline: camera 0 -> buffer 0
    async_copy_b64(lds_addr[0], uvo_base + iclamp);

    F3    kpt  = { 0.0f, 0.0f, 0.0f };
    float best = 0.0f;

    #pragma unroll
    for (int cam = 0; cam < NCAMS; ++cam) {
        // wait for this camera's staged data (only outstanding async op)
        asm volatile("s_wait_asynccnt 0" ::: "memory");
        const float2 uvo = stage[cam & 1][tid];
        // make sure the LDS read retired before the buffer can be overwritten
        asm volatile("s_wait_dscnt 0" ::: "memory");
        if (cam + 1 < NCAMS)   // overlap next camera's load with this camera's compute
            async_copy_b64(lds_addr[(cam + 1) & 1], uvo_base + (long long)(cam + 1) * N + iclamp);

        const float* M = relM[cam];
        const float R00 = M[0], R01 = M[1], R02 = M[2];
        const float R10 = M[3], R11 = M[4], R12 = M[5];
        const float R20 = M[6], R21 = M[7], R22 = M[8];
        const float tx  = M[9], ty  = M[10], tz = M[11];

        // d2 = normalize(R * [(uvo-c)/f, 1])
        const float ax = (uvo.x - cx) * inv_fx;
        const float ay = (uvo.y - cy) * inv_fy;
        F3 d2 = { R00 * ax + R01 * ay + R02,
                  R10 * ax + R11 * ay + R12,
                  R20 * ax + R21 * ay + R22 };
        {
            const float s = __builtin_amdgcn_rsqf(dot3(d2, d2));
            d2.x *= s; d2.y *= s; d2.z *= s;
        }

        const F3 n  = cross3(d1, d2);
        const F3 n2 = cross3(d2, n);
        const float dist = (n2.x * tx + n2.y * ty + n2.z * tz) * __builtin_amdgcn_rcpf(dot3(n2, d1));
        const F3 xyz = { d1.x * dist, d1.y * dist, d1.z * dist };

        // project v through rel camera: p = persp(R^T (v - t)) * f + c   (row-vector (v-t) @ R)
        auto project = [&](float vx, float vy, float vz, float& px, float& py) {
            vx -= tx; vy -= ty; vz -= tz;
            const float wx = vx * R00 + vy * R10 + vz * R20;
            const float wy = vx * R01 + vy * R11 + vz * R21;
            const float wz = vx * R02 + vy * R12 + vz * R22;
            const float iz = __builtin_amdgcn_rcpf(wz);
            px = wx * iz * fx + cx;
            py = wy * iz * fy + cy;
        };

        float p1x, p1y, p2x, p2y, ex, ey;
        project(d1.x,    d1.y,    d1.z,    p1x, p1y);
        project(d1x10.x, d1x10.y, d1x10.z, p2x, p2y);
        project(xyz.x,   xyz.y,   xyz.z,   ex,  ey);

        const float ddx = p1x - p2x, ddy = p1y - p2y;
        const float dis = __builtin_amdgcn_sqrtf(ddx * ddx + ddy * ddy);
        const float edx = ex - uvo.x, edy = ey - uvo.y;
        const float err = __builtin_amdgcn_sqrtf(edx * edx + edy * edy);

        const bool valid = (xyz.z > 1e-6f) & (dis > best) & (err < max_error)
                         & (fminf(uvo.x, uvo.y) > 0.0f);
        kpt.x = valid ? xyz.x : kpt.x;
        kpt.y = valid ? xyz.y : kpt.y;
        kpt.z = valid ? xyz.z : kpt.z;
        best  = valid ? dis   : best;
    }

    if (i < N) {
        const float depth = kpt.z;
        // world-frame: (kpt - Rt[:3,3]) @ Rt[:3,:3]
        const float vx = kpt.x - Rt[3], vy = kpt.y - Rt[7], vz = kpt.z - Rt[11];
        const float ox = vx * Rt[0] + vy * Rt[4] + vz * Rt[8];
        const float oy = vx * Rt[1] + vy * Rt[5] + vz * Rt[9];
        const float oz = vx * Rt[2] + vy * Rt[6] + vz * Rt[10];
        out[i * 3 + 0] = ox;
        out[i * 3 + 1] = oy;
        out[i * 3 + 2] = oz;
        out[3LL * N + i] = depth;
        out[4LL * N + i] = best;
        out[5LL * N + i] = (best > min_dis) ? 1.0f : 0.0f;
    }
}

extern "C" void kernel_launch(void* const* d_in, const int* in_sizes, int n_in,
                              void* d_out, int out_size, void* d_ws, size_t ws_size,
                              hipStream_t stream) {
    const float* uv          = (const float*)d_in[0];
    const float* uvs_others  = (const float*)d_in[1];
    const float* Rt          = (const float*)d_in[2];
    const float* Rts_others  = (const float*)d_in[3];
    const float* fvec        = (const float*)d_in[4];
    const float* centre      = (const float*)d_in[5];
    const float* max_error   = (const float*)d_in[6];
    const float* min_dis     = (const float*)d_in[7];
    float* out = (float*)d_out;

    const int N = in_sizes[0] / 2;
    const int nblocks = (N + BLK - 1) / BLK;

    triangulate_kernel<<<nblocks, BLK, 0, stream>>>(
        uv, uvs_others, Rt, Rts_others, fvec, centre, max_error, min_dis, out, N);
}